// Quantize_28845000360348
// MI455X (gfx1250) — compile-verified
//
#include <hip/hip_runtime.h>

typedef __attribute__((ext_vector_type(2))) float v2f;
typedef __attribute__((ext_vector_type(8))) float v8f;

#define DIM      256
#define NEMB     2048
#define NROWS    32768
#define ROWS_WG  128
#define A_STRIDE 258   // padded LDS row stride (floats)

// ---------- kernel 1: transpose embed [256][2048] -> embedT [2048][256] ----------
__global__ void transpose_kernel(const float* __restrict__ embed, float* __restrict__ embedT) {
  __shared__ float tile[32][33];
  int bx = blockIdx.x;          // k tile 0..63
  int by = blockIdx.y;          // d tile 0..7
  int tx = threadIdx.x;         // 0..31
  int ty = threadIdx.y;         // 0..7
#pragma unroll
  for (int j = 0; j < 4; ++j) {
    int d = by * 32 + ty + j * 8;
    int k = bx * 32 + tx;
    tile[ty + j * 8][tx] = embed[d * NEMB + k];
  }
  __syncthreads();
#pragma unroll
  for (int j = 0; j < 4; ++j) {
    int k = bx * 32 + ty + j * 8;
    int d = by * 32 + tx;
    embedT[k * DIM + d] = tile[tx][ty + j * 8];
  }
}

// ---------- kernel 2: per-code squared norms (coalesced over k) ----------
__global__ void e2_kernel(const float* __restrict__ embed, float* __restrict__ e2) {
  int k = blockIdx.x * blockDim.x + threadIdx.x;
  float s = 0.f;
  for (int d = 0; d < DIM; ++d) { float v = embed[d * NEMB + k]; s += v * v; }
  e2[k] = s;
}

// ---------- kernel 3: main VQ kernel (fp32 WMMA GEMM + argmax + epilogue) ----------
__global__ __launch_bounds__(256) void vq_main_kernel(
    const float* __restrict__ z, const float* __restrict__ embedT,
    const float* __restrict__ e2, float* __restrict__ out_q,
    float* __restrict__ out_ind, float* __restrict__ embed_sumT,
    float* __restrict__ counts, float* __restrict__ diffsum)
{
  extern __shared__ float smem[];
  float* a_lds   = smem;                         // 128*258 floats
  float* b_lds   = smem + ROWS_WG * A_STRIDE;    // 16*256 = 4096 floats (reused for reduction)
  float* idx_lds = b_lds + 16 * DIM;             // 128 floats

  int tid  = threadIdx.x;
  int wave = tid >> 5;
  int lane = tid & 31;
  int row0 = blockIdx.x * ROWS_WG;

  // stage A: 128 rows of z
  for (int i = tid; i < ROWS_WG * DIM; i += 256) {
    int r = i >> 8, k = i & 255;
    a_lds[r * A_STRIDE + k] = z[(row0 + r) * DIM + k];
  }

  int m     = lane & 15;        // A row (within tile) == B column (within tile)
  int khalf = (lane >> 4) * 2;  // lanes 16-31 carry K+2,K+3

  // fixed per-lane LDS bases; all fragment offsets become DS immediates after full unroll
  const float* a_base = &a_lds[(wave * 16 + m) * A_STRIDE + khalf];
  const float* b_base = &b_lds[(khalf >> 1) * 32 + m * 2];

  // staging map: conflict-free ds_store_b64 (lanes 0-15 -> banks 2c,2c+1; 16-31 -> +32)
  int c_s  = tid & 15;          // staged column
  int kp0  = tid >> 4;          // staged k-pair base (0..15)

  float best[8];
  int   bidx[8];
#pragma unroll
  for (int v = 0; v < 8; ++v) { best[v] = -3.4e38f; bidx[v] = 0; }

  // prologue: prefetch B tile 0 + bias 0 into registers
  v2f rb[8];
#pragma unroll
  for (int j = 0; j < 8; ++j)
    rb[j] = *(const v2f*)&embedT[c_s * DIM + (kp0 + j * 16) * 2];
  float bias_next = -0.5f * e2[m];

  __syncthreads();   // A staged

  for (int nt = 0; nt < NEMB / 16; ++nt) {
    __syncthreads();                       // all waves done reading previous B tile
    // dump prefetched B tile into LDS (conflict-free b64 stores)
#pragma unroll
    for (int j = 0; j < 8; ++j)
      *(v2f*)&b_lds[(kp0 + j * 16) * 32 + c_s * 2] = rb[j];

    float bias = bias_next;
    if (nt + 1 < NEMB / 16) {              // uniform branch: prefetch next tile
#pragma unroll
      for (int j = 0; j < 8; ++j)
        rb[j] = *(const v2f*)&embedT[((nt + 1) * 16 + c_s) * DIM + (kp0 + j * 16) * 2];
      bias_next = -0.5f * e2[(nt + 1) * 16 + m];
    }
    __syncthreads();                       // B tile visible

    v8f acc = {0.f, 0.f, 0.f, 0.f, 0.f, 0.f, 0.f, 0.f};
#pragma unroll
    for (int kk = 0; kk < DIM / 4; ++kk) {
      // A: offset kk*16 bytes; B: offset kk*256 bytes — both DS-immediate friendly
      v2f a = *(const v2f*)(a_base + kk * 4);
      v2f b = *(const v2f*)(b_base + kk * 64);
      acc = __builtin_amdgcn_wmma_f32_16x16x4_f32(false, a, false, b, (short)0, acc, false, false);
    }

#pragma unroll
    for (int v = 0; v < 8; ++v) {
      float s = acc[v] + bias;
      if (s > best[v]) { best[v] = s; bidx[v] = nt * 16 + m; }
    }
  }

  __syncthreads();
  // cross-lane argmax reduction via LDS (reuse b_lds: 2048 best + 2048 idx floats)
  float* red_best = b_lds;
  int*   red_idxi = (int*)(b_lds + 2048);
  int hi = lane >> 4;
#pragma unroll
  for (int v = 0; v < 8; ++v) {
    int r = v + 8 * hi;   // C/D layout: lanes 0-15 rows 0-7, lanes 16-31 rows 8-15
    red_best[(wave * 16 + r) * 16 + m] = best[v];
    red_idxi[(wave * 16 + r) * 16 + m] = bidx[v];
  }
  __syncthreads();
  if (tid < ROWS_WG) {
    float bb = -3.4e38f; int bi = 0;
    for (int c = 0; c < 16; ++c) {
      float s = red_best[tid * 16 + c];
      if (s > bb) { bb = s; bi = red_idxi[tid * 16 + c]; }
    }
    idx_lds[tid] = (float)bi;
    out_ind[row0 + tid] = (float)bi;
    atomicAdd(&counts[bi], 1.0f);
  }
  __syncthreads();

  // epilogue: quantize copy (contiguous from embedT), diff partial, scatter-add
  float dsum = 0.f;
  for (int r = 0; r < 16; ++r) {
    int grow = wave * 16 + r;
    int idx  = (int)idx_lds[grow];
    const float* qrow = &embedT[idx * DIM];
    const float* zrow = &a_lds[grow * A_STRIDE];
    int nbase = (row0 + grow) * DIM;
    for (int d = lane; d < DIM; d += 32) {
      float q  = qrow[d];
      float zv = zrow[d];
      out_q[nbase + d] = q;
      float df = q - zv;
      dsum += df * df;
      atomicAdd(&embed_sumT[idx * DIM + d], zv);
    }
  }
  __syncthreads();
  b_lds[tid] = dsum;
  __syncthreads();
  if (tid == 0) {
    float s = 0.f;
    for (int i = 0; i < 256; ++i) s += b_lds[i];
    atomicAdd(diffsum, s);
  }
}

// ---------- kernel 4: EMA cluster size, Laplace smoothing, diff scalar ----------
__global__ void finalize_cs_kernel(const float* __restrict__ cluster_size,
                                   const float* __restrict__ counts,
                                   const float* __restrict__ diffsum,
                                   float* __restrict__ cs, float* __restrict__ out_diff)
{
  __shared__ float red[256];
  __shared__ float nshared;
  int tid = threadIdx.x;
  float local = 0.f;
  float ncs[8];
#pragma unroll
  for (int j = 0; j < 8; ++j) {
    int k = tid * 8 + j;
    ncs[j] = cluster_size[k] * 0.99f + 0.01f * counts[k];
    local += ncs[j];
  }
  red[tid] = local;
  __syncthreads();
  if (tid == 0) {
    float n = 0.f;
    for (int i = 0; i < 256; ++i) n += red[i];
    nshared = n;
    out_diff[0] = diffsum[0] * (1.0f / (float)((long)NROWS * DIM));
  }
  __syncthreads();
  float n = nshared;
  float denom = n + (float)NEMB * 1e-5f;
#pragma unroll
  for (int j = 0; j < 8; ++j) {
    int k = tid * 8 + j;
    cs[k] = (ncs[j] + 1e-5f) / denom * n;
  }
}

// ---------- kernel 5: new_embed = (embed_avg*decay + (1-decay)*embed_sum) / cs ----------
__global__ void new_embed_kernel(const float* __restrict__ embed_avg,
                                 const float* __restrict__ embed_sumT,
                                 const float* __restrict__ cs,
                                 float* __restrict__ out_new_embed)
{
  int i = blockIdx.x * blockDim.x + threadIdx.x;  // 0..524287
  int d = i >> 11;
  int k = i & 2047;
  float na = embed_avg[i] * 0.99f + 0.01f * embed_sumT[k * DIM + d];
  out_new_embed[i] = na / cs[k];
}

extern "C" void kernel_launch(void* const* d_in, const int* in_sizes, int n_in,
                              void* d_out, int out_size, void* d_ws, size_t ws_size,
                              hipStream_t stream) {
  (void)in_sizes; (void)n_in; (void)out_size; (void)ws_size;
  const float* z            = (const float*)d_in[0];
  const float* embed        = (const float*)d_in[1];
  const float* cluster_size = (const float*)d_in[2];
  const float* embed_avg    = (const float*)d_in[3];

  float* out           = (float*)d_out;
  float* out_q         = out;                       // 32768*256
  float* out_diff      = out + 8388608;             // 1
  float* out_ind       = out + 8388609;             // 32768 (as float)
  float* out_new_embed = out + 8421377;             // 256*2048

  float* ws          = (float*)d_ws;
  float* embedT      = ws;                          // 524288
  float* e2          = ws + 524288;                 // 2048
  float* embed_sumT  = ws + 526336;                 // 524288
  float* counts      = ws + 1050624;                // 2048
  float* diffsum     = ws + 1052672;                // 1
  float* cs          = ws + 1052673;                // 2048

  // zero the accumulators (embed_sumT, counts, diffsum are contiguous)
  hipMemsetAsync(embed_sumT, 0, (size_t)(524288 + 2048 + 1) * sizeof(float), stream);

  transpose_kernel<<<dim3(64, 8), dim3(32, 8), 0, stream>>>(embed, embedT);
  e2_kernel<<<8, 256, 0, stream>>>(embed, e2);

  size_t lds_bytes = (size_t)(ROWS_WG * A_STRIDE + 16 * DIM + ROWS_WG) * sizeof(float);
  vq_main_kernel<<<NROWS / ROWS_WG, 256, lds_bytes, stream>>>(
      z, embedT, e2, out_q, out_ind, embed_sumT, counts, diffsum);

  finalize_cs_kernel<<<1, 256, 0, stream>>>(cluster_size, counts, diffsum, cs, out_diff);
  new_embed_kernel<<<2048, 256, 0, stream>>>(embed_avg, embed_sumT, cs, out_new_embed);
}